// DiffusionGraphConvolution_1202590842882
// MI455X (gfx1250) — compile-verified
//
#include <hip/hip_runtime.h>

#define N_NODES 20000
#define N_EDGES 160000
#define BATCH   8
#define D_IN    128
#define D_OUT   128
#define M_FEAT  (BATCH * D_IN)     // 1024 feature rows
#define K_TOT   (5 * D_IN)         // 640
#define M_ROWS  (BATCH * N_NODES)  // 160000 output rows
#define N_CHUNK (K_TOT / 32)       // 20 K-chunks
#define TERM_STRIDE ((size_t)N_NODES * M_FEAT)   // floats between term buffers in ws

typedef __attribute__((ext_vector_type(2))) float v2f;
typedef __attribute__((ext_vector_type(8))) float v8f;

// ---------------------------------------------------------------------------
// 1) Transpose X [1024][20000] -> X0_t [20000][1024] (node-major, coalesced)
// ---------------------------------------------------------------------------
__global__ void transpose_kernel(const float* __restrict__ X, float* __restrict__ Xt) {
    __shared__ float tile[32][33];
    int nx    = blockIdx.x * 32 + threadIdx.x;   // node
    int mBase = blockIdx.y * 32;
    #pragma unroll
    for (int i = 0; i < 32; i += 8) {
        int m = mBase + threadIdx.y + i;
        tile[threadIdx.y + i][threadIdx.x] = X[(size_t)m * N_NODES + nx];
    }
    __syncthreads();
    int m2    = mBase + threadIdx.x;             // feature row
    int nBase = blockIdx.x * 32;
    #pragma unroll
    for (int i = 0; i < 32; i += 8) {
        int nn = nBase + threadIdx.y + i;
        Xt[(size_t)nn * M_FEAT + m2] = tile[threadIdx.x][threadIdx.y + i];
    }
}

// ---------------------------------------------------------------------------
// 2) Permute W: Wp[t*128 + d][o] = W[d*5 + t][o]  (term-major K ordering)
// ---------------------------------------------------------------------------
__global__ void wperm_kernel(const float* __restrict__ W, float* __restrict__ Wp) {
    int idx = blockIdx.x * 256 + threadIdx.x;    // < 640*128 exactly
    int o  = idx & 127;
    int kp = idx >> 7;
    int t  = kp >> 7;
    int d  = kp & 127;
    Wp[idx] = W[(d * 5 + t) * D_OUT + o];
}

// ---------------------------------------------------------------------------
// 3) CSR build: histogram -> single-block scan -> scatter
// ---------------------------------------------------------------------------
__global__ void hist_kernel(const int* __restrict__ rows, int* __restrict__ counts) {
    int e = blockIdx.x * 256 + threadIdx.x;
    if (e < N_EDGES) atomicAdd(&counts[rows[e]], 1);
}

__global__ void scan_kernel(int* __restrict__ cursor /*in: counts, out: row start*/,
                            int* __restrict__ row_ofs) {
    __shared__ int s[1024];
    int t = threadIdx.x;
    const int chunk = 20;                         // 1024*20 >= 20000
    int i0 = t * chunk;
    int i1 = i0 + chunk; if (i1 > N_NODES) i1 = N_NODES;
    if (i0 > N_NODES) i0 = N_NODES;
    int sum = 0;
    for (int i = i0; i < i1; ++i) sum += cursor[i];
    s[t] = sum;
    __syncthreads();
    for (int off = 1; off < 1024; off <<= 1) {
        int v = (t >= off) ? s[t - off] : 0;
        __syncthreads();
        s[t] += v;
        __syncthreads();
    }
    int base = s[t] - sum;                        // exclusive prefix
    for (int i = i0; i < i1; ++i) {
        int c = cursor[i];
        row_ofs[i] = base;
        cursor[i]  = base;                        // scatter cursor
        base += c;
    }
    if (t == 0) row_ofs[N_NODES] = s[1023];
}

__global__ void scatter_kernel(const int* __restrict__ rows, const int* __restrict__ cols,
                               const float* __restrict__ w, int* __restrict__ cursor,
                               int* __restrict__ scol, float* __restrict__ sw) {
    int e = blockIdx.x * 256 + threadIdx.x;
    if (e < N_EDGES) {
        int r = rows[e];
        int p = atomicAdd(&cursor[r], 1);
        scol[p] = cols[e];
        sw[p]   = w[e];
    }
}

// ---------------------------------------------------------------------------
// 4) Gather-form SPMM (node-major): out[i][:] = sum_e w_e * Xin[col_e][:]
//    cheb=1: out = 2*acc - Xsub   (Chebyshev recurrence, fused)
//    One block per node, 256 threads * float4 = 1024 features. No f32 atomics;
//    the 82MB source matrix is L2-resident (192MB L2).
// ---------------------------------------------------------------------------
__global__ void spmm_gather_kernel(const int* __restrict__ row_ofs,
                                   const int* __restrict__ scol,
                                   const float* __restrict__ sw,
                                   const float* __restrict__ Xin,
                                   const float* __restrict__ Xsub,
                                   float* __restrict__ Xout, int cheb) {
    int i  = blockIdx.x;
    int m4 = threadIdx.x * 4;
    int s = row_ofs[i], e = row_ofs[i + 1];
    float4 acc = make_float4(0.f, 0.f, 0.f, 0.f);
    for (int j = s; j < e; ++j) {
        int   c  = scol[j];
        float wv = sw[j];
        const float4 v = *(const float4*)(Xin + (size_t)c * M_FEAT + m4);
        acc.x += wv * v.x; acc.y += wv * v.y; acc.z += wv * v.z; acc.w += wv * v.w;
    }
    if (cheb) {
        const float4 x0 = *(const float4*)(Xsub + (size_t)i * M_FEAT + m4);
        acc.x = 2.f * acc.x - x0.x; acc.y = 2.f * acc.y - x0.y;
        acc.z = 2.f * acc.z - x0.z; acc.w = 2.f * acc.w - x0.w;
    }
    *(float4*)(Xout + (size_t)i * M_FEAT + m4) = acc;
}

// ---------------------------------------------------------------------------
// 5) GEMM: out[160000][128] = A[160000][640] @ Wp[640][128] with
//    V_WMMA_F32_16X16X4_F32 and double-buffered GLOBAL_LOAD_ASYNC_TO_LDS_B128
//    staging (ASYNCcnt-tracked, no VGPR round-trip; chunk i+1 loads overlap
//    chunk i WMMAs). Term buffers sit at uniform stride in ws, so the per-chunk
//    A pointer is pure (branch-free) address arithmetic. Block = 32x128, 8 waves.
// ---------------------------------------------------------------------------
__global__ void gemm_wmma_kernel(const float* __restrict__ terms,   // X0t base; +k*TERM_STRIDE
                                 const float* __restrict__ Wp,
                                 float* __restrict__ out) {
    __shared__ float As[2][32 * 36];    // 32 rows x 32 k, stride 36 (conflict-free, 16B rows)
    __shared__ float Ws[2][32 * 136];   // 32 k x 128 n, stride 136 (half-waves -> disjoint banks)

    int rBase    = blockIdx.x * 32;
    int b        = rBase / N_NODES;       // tile never straddles b (20000 % 32 == 0)
    int nodeBase = rBase % N_NODES;
    int tid  = threadIdx.x;
    int lane = tid & 31;
    int wave = tid >> 5;
    int n0   = wave * 16;

    v8f c0 = {}; v8f c1 = {};

    int arow = tid >> 3;          // 0..31  (A tile row)
    int acg  = (tid & 7) * 4;     // 0..28  (A tile col group)
    int Kb   = (lane >> 4) * 2;   // K sub-pair per half-wave
    int Ml   = lane & 15;         // M index (A) / N index (B)

    // Hoisted per-thread base offsets (loop-invariant).
    const size_t aRowOff = (size_t)(nodeBase + arow) * M_FEAT + b * D_IN + acg;
    const unsigned int laA[2] = { (unsigned int)(size_t)&As[0][arow * 36 + acg],
                                  (unsigned int)(size_t)&As[1][arow * 36 + acg] };

    // Issue one K-chunk's staging as async global->LDS copies (5 x b128/thread).
    auto stage = [&](int ci, int bufsel) {
        // A tile: terms + term*TERM_STRIDE + rowOff + (ci&3)*32  -- branch-free
        const float* ga = terms + (size_t)(ci >> 2) * TERM_STRIDE + aRowOff + (ci & 3) * 32;
        asm volatile("global_load_async_to_lds_b128 %0, %1, off"
                     :: "v"(laA[bufsel]), "v"(ga) : "memory");
        // W tile: linear in ci (chunk ci covers Wp rows ci*32..ci*32+31)
        #pragma unroll
        for (int j = 0; j < 4; ++j) {
            int f    = tid + j * 256;     // 0..1023
            int row  = f >> 5;
            int col4 = (f & 31) * 4;
            const float* gw = Wp + (size_t)ci * (32 * D_OUT) + row * D_OUT + col4;
            unsigned int  lw = (unsigned int)(size_t)&Ws[bufsel][row * 136 + col4];
            asm volatile("global_load_async_to_lds_b128 %0, %1, off"
                         :: "v"(lw), "v"(gw) : "memory");
        }
    };

    stage(0, 0);                          // prologue
    #pragma unroll 2
    for (int ci = 0; ci < N_CHUNK; ++ci) {
        int cur = ci & 1;
        // Only this chunk's async copies are outstanding for this wave.
        asm volatile("s_wait_asynccnt 0x0" ::: "memory");
        __syncthreads();                  // LDS tiles for chunk ci visible block-wide
        if (ci + 1 < N_CHUNK) stage(ci + 1, cur ^ 1);   // overlap with WMMA below
        const float* as  = &As[cur][0];
        const float* wsl = &Ws[cur][0];
        #pragma unroll
        for (int kk = 0; kk < 32; kk += 4) {
            v2f a0, a1, bb;
            a0.x = as[Ml * 36 + kk + Kb];        a0.y = as[Ml * 36 + kk + Kb + 1];
            a1.x = as[(Ml + 16) * 36 + kk + Kb]; a1.y = as[(Ml + 16) * 36 + kk + Kb + 1];
            bb.x = wsl[(kk + Kb) * 136 + n0 + Ml];
            bb.y = wsl[(kk + Kb + 1) * 136 + n0 + Ml];
            c0 = __builtin_amdgcn_wmma_f32_16x16x4_f32(false, a0, false, bb,
                                                       (short)0, c0, false, false);
            c1 = __builtin_amdgcn_wmma_f32_16x16x4_f32(false, a1, false, bb,
                                                       (short)0, c1, false, false);
        }
        __syncthreads();                  // readers done before buffer reuse in ci+2
    }
    // Store: C/D layout = 8 VGPRs; lane<16: M=v, lane>=16: M=v+8; N=lane%16
    int Nl   = lane & 15;
    int half = lane >> 4;
    #pragma unroll
    for (int v = 0; v < 8; ++v) {
        int M0 = v + half * 8;
        out[(size_t)(rBase + M0) * D_OUT + n0 + Nl]      = c0[v];
        out[(size_t)(rBase + 16 + M0) * D_OUT + n0 + Nl] = c1[v];
    }
}

// ---------------------------------------------------------------------------
// Host launcher
// ---------------------------------------------------------------------------
extern "C" void kernel_launch(void* const* d_in, const int* in_sizes, int n_in,
                              void* d_out, int out_size, void* d_ws, size_t ws_size,
                              hipStream_t stream) {
    const float* X     = (const float*)d_in[0];
    const int*   rows1 = (const int*)  d_in[1];
    const int*   cols1 = (const int*)  d_in[2];
    const float* w1    = (const float*)d_in[3];
    const int*   rows2 = (const int*)  d_in[4];
    const int*   cols2 = (const int*)  d_in[5];
    const float* w2    = (const float*)d_in[6];
    const float* W     = (const float*)d_in[7];
    float*       out   = (float*)d_out;

    char* ws = (char*)d_ws;
    const size_t S = TERM_STRIDE * sizeof(float);               // 81.92 MB per term
    float* X0t = (float*)(ws + 0 * S);   // term order must match k' = t*128+d:
    float* X1a = (float*)(ws + 1 * S);   //   [X0, X1a, X2a, X1b, X2b]
    float* X2a = (float*)(ws + 2 * S);
    float* X1b = (float*)(ws + 3 * S);
    float* X2b = (float*)(ws + 4 * S);
    char*  p   = ws + 5 * S;
    float* Wp  = (float*)p;  p += (size_t)K_TOT * D_OUT * sizeof(float);
    int*   ofs1 = (int*)p;   p += (size_t)(N_NODES + 1) * sizeof(int);
    int*   cur1 = (int*)p;   p += (size_t)N_NODES * sizeof(int);
    int*   scol1 = (int*)p;  p += (size_t)N_EDGES * sizeof(int);
    float* sw1 = (float*)p;  p += (size_t)N_EDGES * sizeof(float);
    int*   ofs2 = (int*)p;   p += (size_t)(N_NODES + 1) * sizeof(int);
    int*   cur2 = (int*)p;   p += (size_t)N_NODES * sizeof(int);
    int*   scol2 = (int*)p;  p += (size_t)N_EDGES * sizeof(int);
    float* sw2 = (float*)p;  p += (size_t)N_EDGES * sizeof(float);

    // Transpose X to node-major, permute W to term-major K order
    transpose_kernel<<<dim3(N_NODES / 32, M_FEAT / 32), dim3(32, 8), 0, stream>>>(X, X0t);
    wperm_kernel<<<(K_TOT * D_OUT) / 256, 256, 0, stream>>>(W, Wp);

    // Build CSR for both graphs (int atomics only; no f32 atomics anywhere)
    hipMemsetAsync(cur1, 0, (size_t)N_NODES * sizeof(int), stream);
    hipMemsetAsync(cur2, 0, (size_t)N_NODES * sizeof(int), stream);
    const int EB = (N_EDGES + 255) / 256;
    hist_kernel<<<EB, 256, 0, stream>>>(rows1, cur1);
    hist_kernel<<<EB, 256, 0, stream>>>(rows2, cur2);
    scan_kernel<<<1, 1024, 0, stream>>>(cur1, ofs1);
    scan_kernel<<<1, 1024, 0, stream>>>(cur2, ofs2);
    scatter_kernel<<<EB, 256, 0, stream>>>(rows1, cols1, w1, cur1, scol1, sw1);
    scatter_kernel<<<EB, 256, 0, stream>>>(rows2, cols2, w2, cur2, scol2, sw2);

    // Chebyshev diffusion terms (gather SPMM, L2-resident sources)
    spmm_gather_kernel<<<N_NODES, 256, 0, stream>>>(ofs1, scol1, sw1, X0t, nullptr, X1a, 0);
    spmm_gather_kernel<<<N_NODES, 256, 0, stream>>>(ofs1, scol1, sw1, X1a, X0t, X2a, 1);
    spmm_gather_kernel<<<N_NODES, 256, 0, stream>>>(ofs2, scol2, sw2, X0t, nullptr, X1b, 0);
    spmm_gather_kernel<<<N_NODES, 256, 0, stream>>>(ofs2, scol2, sw2, X1b, X0t, X2b, 1);

    // Final dense projection on the WMMA path
    gemm_wmma_kernel<<<M_ROWS / 32, 256, 0, stream>>>(X0t, Wp, out);
}